// BahdanauAttention_45157286150394
// MI455X (gfx1250) — compile-verified
//
#include <hip/hip_runtime.h>
#include <hip/hip_bf16.h>
#include <math.h>

typedef float v2f __attribute__((ext_vector_type(2)));
typedef float v8f __attribute__((ext_vector_type(8)));

#define B_ 8
#define T_ 128
#define S_ 256
#define H_ 512

// gfx1250 hardware tanh (V_TANH_F32); exp2+rcp fallback.
__device__ __forceinline__ float fast_tanh(float x) {
#if __has_builtin(__builtin_amdgcn_tanhf)
    return __builtin_amdgcn_tanhf(x);
#else
    float cx = fminf(fmaxf(x, -9.2f), 9.2f);
    float e = __builtin_amdgcn_exp2f(cx * 2.885390081777927f); // exp(2x)
    return 1.0f - 2.0f * __builtin_amdgcn_rcpf(e + 1.0f);
#endif
}

__device__ __forceinline__ v8f wmma4(v2f a, v2f b, v8f c) {
    // D = A(16x4,f32) x B(4x16,f32) + C(16x16,f32)
    return __builtin_amdgcn_wmma_f32_16x16x4_f32(
        false, a, false, b, (short)0, c, false, false);
}

// ---- gfx1250 async global->LDS (ASYNCcnt-tracked), with safe fallback ----
#if __has_builtin(__builtin_amdgcn_global_load_async_to_lds_b128) && \
    __has_builtin(__builtin_amdgcn_s_wait_asynccnt)
#define HAVE_ASYNC_LDS 1
// Builtin signature (from hipcc diagnostic): param0 is
//   int __attribute__((vector_size(16))) addrspace(1)*   (global source)
// param1 expected to be the LDS-side pointer of the same vector type.
typedef int async_v4i __attribute__((vector_size(16)));
__device__ __forceinline__ void cp16_async(float* lds_dst, const float* gsrc) {
    __builtin_amdgcn_global_load_async_to_lds_b128(
        (__attribute__((address_space(1))) async_v4i*)(gsrc),
        (__attribute__((address_space(3))) async_v4i*)(lds_dst), 0, 0);
}
__device__ __forceinline__ void cp_async_wait_all() {
    __builtin_amdgcn_s_wait_asynccnt(0);
}
#else
#define HAVE_ASYNC_LDS 0
__device__ __forceinline__ void cp16_async(float* lds_dst, const float* gsrc) {
    *(float4*)lds_dst = *(const float4*)gsrc;
}
__device__ __forceinline__ void cp_async_wait_all() {}
#endif

// C[M,N] = A[M,K] * W[N,K]^T  (nn.Linear, no bias). One 16x16 tile per wave.
// A-frag: lane<16 -> row=lane, cols k0,k0+1 ; lane>=16 -> row=lane-16, cols k0+2,k0+3
// B-frag[k][n] = W[n][k]: lane ln -> W row (nt*16+ln), cols k0+2*(lane>>4) .. +1
__global__ __launch_bounds__(256)
void gemm_wmma(const float* __restrict__ A, const float* __restrict__ W,
               float* __restrict__ C, int M, int N, int K) {
    const int wave = (int)((blockIdx.x * blockDim.x + threadIdx.x) >> 5);
    const int lane = (int)(threadIdx.x & 31);
    const int nTiles = N >> 4;
    const int totTiles = (M >> 4) * nTiles;
    if (wave >= totTiles) return;          // wave-uniform: EXEC stays all-ones

    const int mt = wave / nTiles;
    const int nt = wave - mt * nTiles;
    const int ln = lane & 15;
    const int lh = lane >> 4;              // 0 or 1

    const float* Arow = A + (size_t)(mt * 16 + ln) * K + 2 * lh;
    const float* Wrow = W + (size_t)(nt * 16 + ln) * K + 2 * lh;

    v8f acc = {0.f, 0.f, 0.f, 0.f, 0.f, 0.f, 0.f, 0.f};
#pragma unroll 4
    for (int k = 0; k < K; k += 4) {
        v2f a = *(const v2f*)(Arow + k);
        v2f b = *(const v2f*)(Wrow + k);
        acc = wmma4(a, b, acc);
    }

    const int m0 = mt * 16 + lh * 8;
    const int col = nt * 16 + ln;
#pragma unroll
    for (int i = 0; i < 8; ++i)
        C[(size_t)(m0 + i) * N + col] = acc[i];
}

// Fused: scores (v.tanh(qp+ep)) -> masked softmax -> context = attn x enc.
// One block per (b, 16-row t-tile). 512 threads = 16 waves.
#define EP_STRIDE 516   // 512 + 4 floats pad: float4 reads hit distinct banks
#define SC_STRIDE 260   // 256 + 4 floats pad
__global__ __launch_bounds__(512)
void attn_fused(const float* __restrict__ qp, const float* __restrict__ ep,
                const float* __restrict__ enc, const float* __restrict__ v,
                const int* __restrict__ lens, float* __restrict__ out) {
    extern __shared__ float smem[];
    float* qp_s = smem;                         // 16*512   =  8192 floats
    float* ep_s = qp_s + 16 * H_;               // 32*516   = 16512 floats
    float* v_s  = ep_s + 32 * EP_STRIDE;        // 512 floats
    float* sc_s = v_s + H_;                     // 16*260   =  4160 floats

    const int tid = (int)threadIdx.x;
    const int b   = (int)blockIdx.x >> 3;
    const int mt  = (int)blockIdx.x & 7;
    const int t0  = mt * 16;
    const int len = lens[b];

    // Stage q_proj tile (16 rows contiguous in global) and v into LDS
    // via async global->LDS copies (ASYNCcnt) when available.
    {
        const float* qg = qp + ((size_t)b * T_ + t0) * H_;
        for (int i = tid; i < 16 * (H_ / 4); i += (int)blockDim.x)
            cp16_async(qp_s + i * 4, qg + i * 4);
        for (int i = tid; i < H_ / 4; i += (int)blockDim.x)
            cp16_async(v_s + i * 4, v + i * 4);
    }

    const int t_idx = tid >> 5;   // 0..15
    const int s_idx = tid & 31;   // 0..31
    const float* epb = ep + (size_t)b * S_ * H_;

    // ---- score phase: 8 tiles of 32 source rows staged through LDS ----
    for (int st = 0; st < S_ / 32; ++st) {
        cp_async_wait_all();
        __syncthreads();          // staging visible / previous tile consumed
        for (int i = tid; i < 32 * (H_ / 4); i += (int)blockDim.x) {
            int r = i >> 7;       // /128
            int c = i & 127;
            cp16_async(ep_s + r * EP_STRIDE + c * 4,
                       epb + (size_t)(st * 32 + r) * H_ + c * 4);
        }
        cp_async_wait_all();
        __syncthreads();

        const float4* qrow = (const float4*)(qp_s + t_idx * H_);
        const float4* erow = (const float4*)(ep_s + s_idx * EP_STRIDE);
        const float4* vrow = (const float4*)v_s;
        float acc = 0.f;
#pragma unroll 4
        for (int h = 0; h < H_ / 4; ++h) {
            float4 q = qrow[h], e = erow[h], w = vrow[h];
            acc = fmaf(w.x, fast_tanh(q.x + e.x), acc);
            acc = fmaf(w.y, fast_tanh(q.y + e.y), acc);
            acc = fmaf(w.z, fast_tanh(q.z + e.z), acc);
            acc = fmaf(w.w, fast_tanh(q.w + e.w), acc);
        }
        sc_s[t_idx * SC_STRIDE + st * 32 + s_idx] = acc;
    }
    __syncthreads();

    // ---- masked softmax: one wave per t-row, 8 cols per lane ----
    {
        const int row  = tid >> 5;
        const int lane = tid & 31;
        float* srow = sc_s + row * SC_STRIDE;
        float vals[8];
        float mx = -__builtin_inff();
#pragma unroll
        for (int j = 0; j < 8; ++j) {
            int c = lane * 8 + j;
            float x = (c < len) ? srow[c] : -__builtin_inff();
            vals[j] = x;
            mx = fmaxf(mx, x);
        }
#pragma unroll
        for (int off = 16; off >= 1; off >>= 1)
            mx = fmaxf(mx, __shfl_xor(mx, off, 32));
        float sum = 0.f;
#pragma unroll
        for (int j = 0; j < 8; ++j) {
            int c = lane * 8 + j;
            float e = (c < len)
                ? __builtin_amdgcn_exp2f((vals[j] - mx) * 1.4426950408889634f)
                : 0.f;
            vals[j] = e;
            sum += e;
        }
#pragma unroll
        for (int off = 16; off >= 1; off >>= 1)
            sum += __shfl_xor(sum, off, 32);
        float inv = 1.0f / sum;
#pragma unroll
        for (int j = 0; j < 8; ++j) srow[lane * 8 + j] = vals[j] * inv;
    }
    __syncthreads();

    // ---- context: D(16x512) = attn(16x256) x enc_b(256x512) via WMMA ----
    {
        const int wave = tid >> 5;
        const int lane = tid & 31;
        const int lh = lane >> 4;
        const int ln = lane & 15;
        const float* encb = enc + (size_t)b * S_ * H_;
#pragma unroll
        for (int i = 0; i < 2; ++i) {               // 16 waves x 2 n-tiles = 512 cols
            const int nt = wave * 2 + i;
            const int n  = nt * 16 + ln;
            const float* arow = sc_s + ln * SC_STRIDE + 2 * lh;
            v8f acc = {0.f, 0.f, 0.f, 0.f, 0.f, 0.f, 0.f, 0.f};
#pragma unroll 4
            for (int k = 0; k < S_; k += 4) {
                v2f a = *(const v2f*)(arow + k);
                v2f bb;
                bb.x = encb[(size_t)(k + 2 * lh) * H_ + n];
                bb.y = encb[(size_t)(k + 2 * lh + 1) * H_ + n];
                acc = wmma4(a, bb, acc);
            }
            float* orow = out + ((size_t)b * T_ + t0 + lh * 8) * H_ + n;
#pragma unroll
            for (int r = 0; r < 8; ++r) orow[(size_t)r * H_] = acc[r];
        }
    }
}

extern "C" void kernel_launch(void* const* d_in, const int* in_sizes, int n_in,
                              void* d_out, int out_size, void* d_ws, size_t ws_size,
                              hipStream_t stream) {
    const float* query = (const float*)d_in[0];   // (B,T,H)
    const float* enc   = (const float*)d_in[1];   // (B,S,H)
    const int*   lens  = (const int*)d_in[2];     // (B,)
    const float* Ws    = (const float*)d_in[3];   // (H,H)
    const float* Wh    = (const float*)d_in[4];   // (H,H)
    const float* v     = (const float*)d_in[5];   // (H,)
    float* out = (float*)d_out;

    float* qp = (float*)d_ws;                       // (B*T, H) = 2 MB
    float* ep = qp + (size_t)B_ * T_ * H_;          // (B*S, H) = 4 MB

    // q_proj: (1024x512)x(512x512)^T -> 64*32 tiles -> 2048 waves -> 256 blocks
    gemm_wmma<<<256, 256, 0, stream>>>(query, Ws, qp, B_ * T_, H_, H_);
    // e_proj: (2048x512)x(512x512)^T -> 128*32 tiles -> 4096 waves -> 512 blocks
    gemm_wmma<<<512, 256, 0, stream>>>(enc, Wh, ep, B_ * S_, H_, H_);

    const size_t smem_bytes =
        (size_t)(16 * H_ + 32 * EP_STRIDE + H_ + 16 * SC_STRIDE) * sizeof(float);
    attn_fused<<<dim3(B_ * (T_ / 16)), 512, smem_bytes, stream>>>(
        qp, ep, enc, v, lens, out);
}